// R_GCN_79173427135060
// MI455X (gfx1250) — compile-verified
//
#include <hip/hip_runtime.h>

#define N_NODES 50000
#define N_EDGES 800000
#define N_ENTS  200000
#define N_RELS  500
#define DIM     128
#define LDS_STRIDE 132   // pad 128 -> 132 floats: column reads hit distinct banks

typedef __attribute__((ext_vector_type(2))) float v2f;
typedef __attribute__((ext_vector_type(8))) float v8f;

// ---------------- zero int workspace (degree histogram) ---------------------
__global__ void zero_int_kernel(int* __restrict__ p, int n) {
  int i = blockIdx.x * blockDim.x + threadIdx.x;
  int stride = gridDim.x * blockDim.x;
  for (; i < n; i += stride) p[i] = 0;
}

// ---------------- h = prev_ent_embs[node_id]  (float4 coalesced) ------------
__global__ void gather_h_kernel(const float* __restrict__ ent,
                                const int* __restrict__ node_id,
                                float* __restrict__ h) {
  int t = blockIdx.x * blockDim.x + threadIdx.x;
  const int total = N_NODES * (DIM / 4);
  if (t >= total) return;
  int node = t >> 5;        // DIM/4 == 32
  int c4   = t & 31;
  int e = node_id[node];
  reinterpret_cast<float4*>(h)[(size_t)node * 32 + c4] =
      reinterpret_cast<const float4*>(ent)[(size_t)e * 32 + c4];
}

// ---------------- in-degree histogram (1 int atomic per edge) ---------------
__global__ void deg_count_kernel(const int* __restrict__ dst,
                                 int* __restrict__ degi) {
  int e = blockIdx.x * blockDim.x + threadIdx.x;
  if (e < N_EDGES)
    __hip_atomic_fetch_add(&degi[dst[e]], 1, __ATOMIC_RELAXED,
                           __HIP_MEMORY_SCOPE_AGENT);
}

// ---------------- single-block exclusive scan over degrees ------------------
// 1024 threads (32 waves), Hillis-Steele per 1024-chunk + running carry.
__global__ __launch_bounds__(1024)
void scan_kernel(const int* __restrict__ degi,
                 int* __restrict__ offs,
                 int* __restrict__ cursor) {
  __shared__ int buf[1024];
  __shared__ int carry;
  if (threadIdx.x == 0) carry = 0;
  __syncthreads();
  for (int base = 0; base < N_NODES; base += 1024) {
    int i = base + threadIdx.x;
    int v = (i < N_NODES) ? degi[i] : 0;
    buf[threadIdx.x] = v;
    __syncthreads();
    for (int off = 1; off < 1024; off <<= 1) {
      int t = (threadIdx.x >= (unsigned)off) ? buf[threadIdx.x - off] : 0;
      __syncthreads();
      buf[threadIdx.x] += t;
      __syncthreads();
    }
    int excl = carry + buf[threadIdx.x] - v;   // exclusive prefix
    if (i < N_NODES) { offs[i] = excl; cursor[i] = excl; }
    __syncthreads();
    if (threadIdx.x == 0) carry += buf[1023];
    __syncthreads();
  }
}

// ---------------- CSR edge-index scatter (1 int atomic per edge) ------------
__global__ void scatter_eidx_kernel(const int* __restrict__ dst,
                                    int* __restrict__ cursor,
                                    int* __restrict__ eidx) {
  int e = blockIdx.x * blockDim.x + threadIdx.x;
  if (e < N_EDGES) {
    int pos = __hip_atomic_fetch_add(&cursor[dst[e]], 1, __ATOMIC_RELAXED,
                                     __HIP_MEMORY_SCOPE_AGENT);
    eidx[pos] = e;
  }
}

// ------- pull-aggregate: agg_pre[n] = sum_in (h[src] + rel[etype]) ----------
// one wave32 per dst node; lane owns a float4 column slice; zero atomics.
__global__ void aggregate_kernel(const float* __restrict__ h,
                                 const float* __restrict__ rel,
                                 const int* __restrict__ src,
                                 const int* __restrict__ etype,
                                 const int* __restrict__ eidx,
                                 const int* __restrict__ offs,
                                 const int* __restrict__ degi,
                                 float* __restrict__ agg) {
  int gid  = blockIdx.x * blockDim.x + threadIdx.x;
  int node = gid >> 5;
  int lane = gid & 31;
  if (node >= N_NODES) return;
  int start = offs[node];
  int cnt   = degi[node];
  float4 acc = make_float4(0.f, 0.f, 0.f, 0.f);
  for (int k = 0; k < cnt; ++k) {
    int e = eidx[start + k];            // wave-uniform -> broadcast load
    int s = src[e], r = etype[e];
    float4 hv = reinterpret_cast<const float4*>(h   + (size_t)s * DIM)[lane];
    float4 rv = reinterpret_cast<const float4*>(rel + (size_t)r * DIM)[lane];
    acc.x += hv.x + rv.x;
    acc.y += hv.y + rv.y;
    acc.z += hv.z + rv.z;
    acc.w += hv.w + rv.w;
  }
  reinterpret_cast<float4*>(agg + (size_t)node * DIM)[lane] = acc;
}

// -------- out = relu(agg_pre@Wn * norm + h@(has_in ? Wl : We)) --------------
// block = 8 waves, one 16-row tile of nodes; wave w -> column tile w*16.
// V_WMMA_F32_16X16X4_F32: A 16x4 (lane=M, khalf per half-wave), B 4x16,
// C/D 8 VGPRs (VGPR j: M=j / M=j+8 per half-wave, N=lane&15).
__global__ __launch_bounds__(256)
void finalize_wmma_kernel(const float* __restrict__ agg,
                          const float* __restrict__ h,
                          const int* __restrict__ degi,
                          const float* __restrict__ Wn,
                          const float* __restrict__ Wl,
                          const float* __restrict__ We,
                          float* __restrict__ out) {
  __shared__ float sAgg[16 * LDS_STRIDE];
  __shared__ float sH[16 * LDS_STRIDE];
  __shared__ int   sDeg[16];

  const int row_base = blockIdx.x * 16;
  for (int idx = threadIdx.x; idx < 16 * DIM; idx += 256) {
    int rr = idx >> 7, cc = idx & 127;
    size_t g = (size_t)(row_base + rr) * DIM + cc;
    sAgg[rr * LDS_STRIDE + cc] = agg[g];
    sH[rr * LDS_STRIDE + cc]   = h[g];
  }
  if (threadIdx.x < 16) sDeg[threadIdx.x] = degi[row_base + threadIdx.x];
  __syncthreads();

  const int wave   = threadIdx.x >> 5;
  const int lane   = threadIdx.x & 31;
  const int n_base = wave * 16;
  const int m      = lane & 15;          // also the N index for B/D
  const int khalf  = (lane >> 4) << 1;   // 0 or 2

  v8f accN = {}, accL = {}, accE = {};
  for (int k0 = 0; k0 < DIM; k0 += 4) {
    int kk = k0 + khalf;
    v2f aA, aH, bN, bL, bE;
    aA.x = sAgg[m * LDS_STRIDE + kk];  aA.y = sAgg[m * LDS_STRIDE + kk + 1];
    aH.x = sH[m * LDS_STRIDE + kk];    aH.y = sH[m * LDS_STRIDE + kk + 1];
    int wi = kk * DIM + n_base + m;    // W[kk][n], row-major DxD
    bN.x = Wn[wi]; bN.y = Wn[wi + DIM];
    bL.x = Wl[wi]; bL.y = Wl[wi + DIM];
    bE.x = We[wi]; bE.y = We[wi + DIM];
    accN = __builtin_amdgcn_wmma_f32_16x16x4_f32(false, aA, false, bN, (short)0, accN, false, false);
    accL = __builtin_amdgcn_wmma_f32_16x16x4_f32(false, aH, false, bL, (short)0, accL, false, false);
    accE = __builtin_amdgcn_wmma_f32_16x16x4_f32(false, aH, false, bE, (short)0, accE, false, false);
  }

  const int mhi = (lane >> 4) << 3;      // 0 or 8
  for (int j = 0; j < 8; ++j) {
    int mr = j + mhi;
    int   dv    = sDeg[mr];
    float norm  = dv > 0 ? 1.0f / (float)dv : 0.0f;
    float loopv = dv > 0 ? accL[j] : accE[j];
    float val   = accN[j] * norm + loopv;
    out[(size_t)(row_base + mr) * DIM + n_base + m] = val > 0.0f ? val : 0.0f;
  }
}

extern "C" void kernel_launch(void* const* d_in, const int* in_sizes, int n_in,
                              void* d_out, int out_size, void* d_ws, size_t ws_size,
                              hipStream_t stream) {
  const float* prev_ent = (const float*)d_in[0];
  const float* rel      = (const float*)d_in[1];
  const float* Wl       = (const float*)d_in[2];   // loop_weight
  const float* We       = (const float*)d_in[3];   // evolve_loop_weight
  const float* Wn       = (const float*)d_in[4];   // weight_neighbor
  const int*   node_id  = (const int*)d_in[5];
  const int*   src      = (const int*)d_in[6];
  const int*   dst      = (const int*)d_in[7];
  const int*   etype    = (const int*)d_in[8];
  float* out = (float*)d_out;

  // workspace layout (all 4-byte types, contiguous)
  float* h      = (float*)d_ws;                    // [N_NODES*DIM]
  float* agg    = h + (size_t)N_NODES * DIM;       // [N_NODES*DIM] fully overwritten
  int*   degi   = (int*)(agg + (size_t)N_NODES * DIM); // [N_NODES]
  int*   offs   = degi + N_NODES;                  // [N_NODES]
  int*   cursor = offs + N_NODES;                  // [N_NODES]
  int*   eidx   = cursor + N_NODES;                // [N_EDGES]

  // 1) zero degree histogram
  zero_int_kernel<<<256, 256, 0, stream>>>(degi, N_NODES);

  // 2) gather node features
  {
    int total = N_NODES * (DIM / 4);
    gather_h_kernel<<<(total + 255) / 256, 256, 0, stream>>>(prev_ent, node_id, h);
  }

  // 3) in-degree histogram
  deg_count_kernel<<<(N_EDGES + 255) / 256, 256, 0, stream>>>(dst, degi);

  // 4) exclusive scan -> CSR row offsets (+ scatter cursors)
  scan_kernel<<<1, 1024, 0, stream>>>(degi, offs, cursor);

  // 5) CSR edge-index scatter
  scatter_eidx_kernel<<<(N_EDGES + 255) / 256, 256, 0, stream>>>(dst, cursor, eidx);

  // 6) pull-mode aggregation (no float atomics)
  {
    long long threads = (long long)N_NODES * 32;
    int blocks = (int)((threads + 255) / 256);
    aggregate_kernel<<<blocks, 256, 0, stream>>>(h, rel, src, etype, eidx, offs, degi, agg);
  }

  // 7) fused triple-GEMM + norm + select + relu via f32 WMMA
  finalize_wmma_kernel<<<N_NODES / 16, 256, 0, stream>>>(agg, h, degi, Wn, Wl, We, out);
}